// RPN_14851996909759
// MI455X (gfx1250) — compile-verified
//
#include <hip/hip_runtime.h>
#include <cstdint>
#include <cstddef>

#define B_ 8
#define A_ 3
#define H_ 200
#define W_ 304
#define N_ (A_ * H_ * W_)   // 182400
#define K_ 2000
#define MWORDS 64           // ceil(2000/32)=63, padded to 64
#define CAP 4096
#define LOG_MAX_ 4.135166556742356f
#define NMS_THR_ 0.7f
#define MIN_SIZE_ 16.0f

// Monotonic uint key: larger float -> larger uint.
__device__ __forceinline__ uint32_t fkey(float f) {
  uint32_t u = __float_as_uint(f);
  uint32_t m = (u & 0x80000000u) ? 0xFFFFFFFFu : 0x80000000u;
  return u ^ m;
}
__device__ __forceinline__ float key_to_float(uint32_t k) {
  uint32_t u = (k & 0x80000000u) ? (k ^ 0x80000000u) : ~k;
  return __uint_as_float(u);
}
// LDS byte offset of a __shared__ object (low 32 bits of its generic address).
__device__ __forceinline__ unsigned lds_off(const void* p) {
  return (unsigned)(size_t)p;
}

// ---------------------------------------------------------------------------
// Kernel 1: per-batch exact top-2000 (radix-select + bitonic sort), sorted
// descending with stable (lower-index-first) tie-break, matching lax.top_k.
// Writes reference-flattened indices i = (h*W+w)*A + a and the scores.
// ---------------------------------------------------------------------------
__global__ __launch_bounds__(1024) void topk_kernel(
    const float* __restrict__ cls, uint32_t* __restrict__ idx_out,
    float* __restrict__ score_out) {
  __shared__ uint32_t hist[256];
  __shared__ unsigned long long cand[CAP];
  __shared__ int s_target;
  __shared__ uint32_t s_bin, s_cnt;

  const int b = blockIdx.x;
  const int tid = threadIdx.x;
  const float* __restrict__ sb = cls + (size_t)b * N_;

  if (tid == 0) { s_target = K_; s_cnt = 0; }
  __syncthreads();

  uint32_t prefix = 0, pmask = 0;
  for (int pass = 0; pass < 4; ++pass) {
    const int shift = 24 - 8 * pass;
    for (int i = tid; i < 256; i += 1024) hist[i] = 0;
    __syncthreads();
    for (int t = tid; t < N_; t += 1024) {
      if (pass == 0 && t + 16384 < N_)
        __builtin_prefetch(sb + t + 16384, 0, 0);  // global_prefetch_b8
      uint32_t k = fkey(sb[t]);
      if ((k & pmask) == prefix)
        atomicAdd(&hist[(k >> shift) & 0xFFu], 1u);
    }
    __syncthreads();
    if (tid == 0) {
      int tgt = s_target, cum = 0;
      uint32_t bin = 255;
      for (;;) {
        int c = (int)hist[bin];
        if (cum + c >= tgt || bin == 0) break;
        cum += c;
        --bin;
      }
      s_bin = bin;
      s_target = tgt - cum;
    }
    __syncthreads();
    prefix |= s_bin << shift;
    pmask |= 0xFFu << shift;
    __syncthreads();
  }
  const uint32_t T = prefix;  // key of the K-th largest element

  // Compact all candidates with key >= T (count ~= 2000 barring mass ties).
  for (int t = tid; t < N_; t += 1024) {
    uint32_t k = fkey(sb[t]);
    if (k >= T) {
      uint32_t slot = atomicAdd(&s_cnt, 1u);
      if (slot < CAP) {
        uint32_t a = (uint32_t)t / (H_ * W_);
        uint32_t r = (uint32_t)t % (H_ * W_);
        uint32_t i = r * A_ + a;  // reference flattened index
        cand[slot] =
            ((unsigned long long)k << 32) | (uint32_t)(0xFFFFFFFFu - i);
      }
    }
  }
  __syncthreads();
  uint32_t cnt = s_cnt;
  if (cnt > CAP) cnt = CAP;
  for (int t = tid; t < CAP; t += 1024)
    if ((uint32_t)t >= cnt) cand[t] = 0ull;
  __syncthreads();

  // Bitonic sort, descending (key desc; for equal keys, smaller idx first).
  for (unsigned kk = 2; kk <= CAP; kk <<= 1) {
    for (unsigned js = kk >> 1; js > 0; js >>= 1) {
      for (unsigned t = tid; t < CAP; t += 1024) {
        unsigned ixj = t ^ js;
        if (ixj > t) {
          unsigned long long va = cand[t], vb = cand[ixj];
          bool desc = ((t & kk) == 0);
          if (desc ? (va < vb) : (va > vb)) {
            cand[t] = vb;
            cand[ixj] = va;
          }
        }
      }
      __syncthreads();
    }
  }

  for (int t = tid; t < K_; t += 1024) {
    unsigned long long v = cand[t];
    uint32_t k = (uint32_t)(v >> 32);
    uint32_t i = 0xFFFFFFFFu - (uint32_t)(v & 0xFFFFFFFFu);
    idx_out[(size_t)b * K_ + t] = i;
    score_out[(size_t)b * K_ + t] = key_to_float(k);
  }
}

// ---------------------------------------------------------------------------
// Kernel 2: gather deltas/anchors at top-k indices, decode boxes, validity.
// deltas[c] = regs[b, 4a+c, h, w]; anchors gathered from input directly.
// ---------------------------------------------------------------------------
__global__ __launch_bounds__(256) void decode_kernel(
    const float* __restrict__ regs, const float* __restrict__ anchors,
    const uint32_t* __restrict__ idx_ws, const int* __restrict__ img_h,
    const int* __restrict__ img_w, float* __restrict__ boxes,
    uint32_t* __restrict__ valid) {
  int e = blockIdx.x * 256 + threadIdx.x;
  if (e >= B_ * K_) return;
  int b = e / K_;
  uint32_t i = idx_ws[e];
  uint32_t a = i % A_;
  uint32_t pos = i / A_;
  uint32_t h = pos / W_;
  uint32_t w = pos % W_;
  const size_t plane = (size_t)H_ * W_;
  const float* rb =
      regs + ((size_t)b * (4 * A_) + 4 * a) * plane + (size_t)h * W_ + w;
  float dx = rb[0];
  float dy = rb[plane];
  float dh = fminf(rb[2 * plane], LOG_MAX_);
  float dw = fminf(rb[3 * plane], LOG_MAX_);
  const float* anc = anchors + ((size_t)b * N_ + i) * 4;
  float ah = anc[3] - anc[1];
  float aw = anc[2] - anc[0];
  float px = aw * 0.5f + dx * aw;  // note: reference uses aw/2, ah/2 (not ctr)
  float py = ah * 0.5f + dy * ah;
  float ph = expf(dh) * ah;
  float pw = expf(dw) * aw;
  float x1 = px - pw * 0.5f, y1 = py - ph * 0.5f;
  float x2 = px + pw * 0.5f, y2 = py + ph * 0.5f;
  ((float4*)boxes)[e] = make_float4(x1, y1, x2, y2);
  float wi = (float)img_w[0], hi = (float)img_h[0];
  float bw = fminf(fmaxf(x2, 0.f), wi) - fminf(fmaxf(x1, 0.f), wi);
  float bh = fminf(fmaxf(y2, 0.f), hi) - fminf(fmaxf(y1, 0.f), hi);
  valid[e] = (bw >= MIN_SIZE_ && bh >= MIN_SIZE_) ? 1u : 0u;
}

// ---------------------------------------------------------------------------
// Kernel 3: suppression bit-matrix. Block = 128 rows x 128-col tile. Column
// boxes staged global->LDS via the CDNA5 async DMA path (ASYNCcnt).
// mask[b][i][w] bit j set  <=>  j > i and IoU(i,j) > 0.7.
// ---------------------------------------------------------------------------
__global__ __launch_bounds__(128) void nms_mask_kernel(
    const float* __restrict__ boxes, uint32_t* __restrict__ mask) {
  __shared__ __align__(16) float tile[128 * 4];
  const int b = blockIdx.z;
  const int row0 = blockIdx.x * 128;
  const int col0 = blockIdx.y * 128;
  const int tid = threadIdx.x;

  {
    int j = col0 + tid;
    int jc = (j < K_) ? j : (K_ - 1);  // clamp: all lanes DMA in-bounds
    unsigned lo = lds_off(tile) + (unsigned)tid * 16u;
    unsigned long long ga =
        (unsigned long long)(size_t)(boxes + ((size_t)b * K_ + jc) * 4);
    asm volatile("global_load_async_to_lds_b128 %0, %1, off" ::"v"(lo),
                 "v"(ga)
                 : "memory");
    asm volatile("s_wait_asynccnt 0x0" ::: "memory");
  }
  __syncthreads();

  const int row = row0 + tid;
  if (row < K_) {
    float4 r = ((const float4*)boxes)[(size_t)b * K_ + row];
    float rarea = (r.z - r.x) * (r.w - r.y);
    uint32_t wds[4];
#pragma unroll
    for (int g = 0; g < 4; ++g) {
      uint32_t wbits = 0;
      for (int k = 0; k < 32; ++k) {
        int jj = g * 32 + k;
        int j = col0 + jj;
        if (j < K_ && j > row) {
          float4 c = ((const float4*)tile)[jj];
          float iw = fmaxf(fminf(r.z, c.z) - fmaxf(r.x, c.x), 0.f);
          float ih = fmaxf(fminf(r.w, c.w) - fmaxf(r.y, c.y), 0.f);
          float inter = iw * ih;
          float carea = (c.z - c.x) * (c.w - c.y);
          float uni = fmaxf(rarea + carea - inter, 1e-6f);
          if (inter > NMS_THR_ * uni) wbits |= 1u << k;
        }
      }
      wds[g] = wbits;
    }
    uint32_t* mrow = mask + ((size_t)b * K_ + row) * MWORDS + (col0 >> 5);
    mrow[0] = wds[0];
    mrow[1] = wds[1];
    mrow[2] = wds[2];
    mrow[3] = wds[3];
  }
}

// ---------------------------------------------------------------------------
// Kernel 4: sequential greedy scan (exactly the reference recurrence: only
// rows that are still kept at step i suppress). The 2000-step chain is pure
// L2-latency; hide it by double-buffering mask rows into LDS with the async
// DMA engine: row i+1 streams in while row i is being ANDed.
// ---------------------------------------------------------------------------
__global__ __launch_bounds__(64) void nms_reduce_kernel(
    const float* __restrict__ boxes, const uint32_t* __restrict__ valid,
    const uint32_t* __restrict__ mask, float* __restrict__ out_boxes,
    float* __restrict__ out_keep) {
  __shared__ uint32_t keepw[MWORDS];
  __shared__ __align__(16) uint32_t rowbuf[2][MWORDS];
  const int b = blockIdx.x, tid = threadIdx.x;
  {
    uint32_t wbits = 0;
    for (int k = 0; k < 32; ++k) {
      int e = tid * 32 + k;
      if (e < K_ && valid[(size_t)b * K_ + e]) wbits |= 1u << k;
    }
    keepw[tid] = wbits;
  }
  const uint32_t* mb = mask + (size_t)b * K_ * MWORDS;

  // Prime the pipeline: async-load mask row 0 (each lane one word).
  {
    unsigned lo = lds_off(rowbuf[0]) + (unsigned)tid * 4u;
    unsigned long long ga = (unsigned long long)(size_t)(mb + tid);
    asm volatile("global_load_async_to_lds_b32 %0, %1, off" ::"v"(lo), "v"(ga)
                 : "memory");
  }
  __syncthreads();

  for (int i = 0; i < K_; ++i) {
    asm volatile("s_wait_asynccnt 0x0" ::: "memory");  // row i resident
    __syncthreads();
    if (i + 1 < K_) {  // overlap: fetch row i+1 during this row's update
      unsigned lo = lds_off(rowbuf[(i + 1) & 1]) + (unsigned)tid * 4u;
      unsigned long long ga =
          (unsigned long long)(size_t)(mb + (size_t)(i + 1) * MWORDS + tid);
      asm volatile("global_load_async_to_lds_b32 %0, %1, off" ::"v"(lo),
                   "v"(ga)
                   : "memory");
    }
    uint32_t alive = keepw[i >> 5];  // row i never holds its own bit (j > i)
    if ((alive >> (i & 31)) & 1u) {
      keepw[tid] &= ~rowbuf[i & 1][tid];
    }
    __syncthreads();
  }

  for (int e = tid; e < K_; e += 64) {
    float m = ((keepw[e >> 5] >> (e & 31)) & 1u) ? 1.0f : 0.0f;
    size_t o = (size_t)b * K_ + e;
    float4 bx = ((const float4*)boxes)[o];
    ((float4*)out_boxes)[o] =
        make_float4(bx.x * m, bx.y * m, bx.z * m, bx.w * m);
    out_keep[o] = m;
  }
}

// ---------------------------------------------------------------------------
extern "C" void kernel_launch(void* const* d_in, const int* in_sizes, int n_in,
                              void* d_out, int out_size, void* d_ws,
                              size_t ws_size, hipStream_t stream) {
  (void)in_sizes; (void)n_in; (void)out_size; (void)ws_size;
  const float* cls = (const float*)d_in[0];
  const float* regs = (const float*)d_in[1];
  const float* anchors = (const float*)d_in[2];
  const int* img_h = (const int*)d_in[3];
  const int* img_w = (const int*)d_in[4];

  float* out = (float*)d_out;
  float* out_boxes = out;                          // B*K*4 = 64000
  float* out_scores = out + (size_t)B_ * K_ * 4;   // B*K   = 16000
  float* out_keep = out + (size_t)B_ * K_ * 5;     // B*K   = 16000

  char* ws = (char*)d_ws;
  uint32_t* idx_ws = (uint32_t*)ws;                       // 64000 B
  float* boxes_ws = (float*)(ws + 64000);                 // 256000 B (16-aligned)
  uint32_t* valid_ws = (uint32_t*)(ws + 320000);          // 64000 B
  uint32_t* mask_ws = (uint32_t*)(ws + 384000);           // 8*2000*64*4 = 4.096 MB

  hipLaunchKernelGGL(topk_kernel, dim3(B_), dim3(1024), 0, stream, cls, idx_ws,
                     out_scores);
  hipLaunchKernelGGL(decode_kernel, dim3((B_ * K_ + 255) / 256), dim3(256), 0,
                     stream, regs, anchors, idx_ws, img_h, img_w, boxes_ws,
                     valid_ws);
  hipLaunchKernelGGL(nms_mask_kernel, dim3(16, 16, B_), dim3(128), 0, stream,
                     boxes_ws, mask_ws);
  hipLaunchKernelGGL(nms_reduce_kernel, dim3(B_), dim3(64), 0, stream, boxes_ws,
                     valid_ws, mask_ws, out_boxes, out_keep);
}